// DualGCN_20358144983303
// MI455X (gfx1250) — compile-verified
//
#include <hip/hip_runtime.h>

typedef __attribute__((ext_vector_type(2))) float v2f;
typedef __attribute__((ext_vector_type(8))) float v8f;

// ---------------- degree / norm ----------------

__global__ void deg_init_kernel(float* __restrict__ deg, int n) {
    int i = blockIdx.x * blockDim.x + threadIdx.x;
    if (i < n) deg[i] = 1.0f;                 // self-loop contributes 1
}

__global__ void deg_accum_kernel(const int* __restrict__ dst, float* __restrict__ deg, int E) {
    int e = blockIdx.x * blockDim.x + threadIdx.x;
    if (e < E) atomicAdd(&deg[dst[e]], 1.0f);
}

__global__ void rsqrt_kernel(float* __restrict__ d, int n) {
    int i = blockIdx.x * blockDim.x + threadIdx.x;
    if (i < n) d[i] = rsqrtf(d[i]);           // deg >= 1 always (self-loops)
}

// ---------------- dense GEMM via V_WMMA_F32_16X16X4_F32 ----------------
// H[n_rows, n_cols] = X[n_rows, k_dim] @ W[k_dim, n_cols]  (all row-major fp32)
// One wave computes one 16x16 tile; K stepped by 4. Requires n_rows%16==0,
// n_cols%16==0, k_dim%4==0 (true here: 50000, 96/32, 96).
__global__ void gemm_wmma_f32(const float* __restrict__ X, const float* __restrict__ W,
                              float* __restrict__ H, int n_rows, int k_dim, int n_cols) {
    const int wave = threadIdx.x >> 5;
    const int lane = threadIdx.x & 31;
    const int tiles_per_row = n_cols >> 4;
    const int total_tiles   = (n_rows >> 4) * tiles_per_row;
    const int tile = blockIdx.x * (blockDim.x >> 5) + wave;
    if (tile >= total_tiles) return;          // wave-uniform: EXEC all-1s for WMMA

    const int rt = tile / tiles_per_row;
    const int ct = tile - rt * tiles_per_row;
    const int r  = lane & 15;
    const int hi = lane >> 4;

    const float* __restrict__ arow = X + (size_t)(rt * 16 + r) * k_dim;
    const int bcol = ct * 16 + r;

    v8f acc = {};
    for (int k = 0; k < k_dim; k += 4) {
        v2f a, b;
        // A 16x4 layout: lane r holds K = k+2*hi (v0) and K = k+2*hi+1 (v1)
        a.x = arow[k + 2 * hi];
        a.y = arow[k + 2 * hi + 1];
        // B 4x16 layout mirrors A with N in place of M
        b.x = W[(size_t)(k + 2 * hi)     * n_cols + bcol];
        b.y = W[(size_t)(k + 2 * hi + 1) * n_cols + bcol];
        acc = __builtin_amdgcn_wmma_f32_16x16x4_f32(
            /*neg_a=*/false, a, /*neg_b=*/false, b,
            /*c_mod=*/(short)0, acc, /*reuse_a=*/false, /*reuse_b=*/false);
    }

    // D layout: VGPR v -> M = v + 8*hi, N = r
#pragma unroll
    for (int v = 0; v < 8; ++v) {
        H[(size_t)(rt * 16 + v + 8 * hi) * n_cols + ct * 16 + r] = acc[v];
    }
}

// ---------------- aggregation ----------------

// out[i,f] = bias[f] + dinv[i]^2 * h[i,f]   (self-loop term; also fully initializes out)
__global__ void init_out_kernel(const float* __restrict__ h, const float* __restrict__ dinv,
                                const float* __restrict__ bias, float* __restrict__ out,
                                int n, int f) {
    int idx = blockIdx.x * blockDim.x + threadIdx.x;
    if (idx >= n * f) return;
    int i = idx / f;
    int j = idx - i * f;
    float di = dinv[i];
    out[idx] = bias[j] + h[idx] * di * di;
}

// out[dst,f] += dinv[src]*dinv[dst] * h[src,f], 4 features per thread
__global__ void scatter_kernel(const float* __restrict__ h, const float* __restrict__ dinv,
                               const int* __restrict__ src, const int* __restrict__ dst,
                               float* __restrict__ out, int E, int f) {
    const int gpe = f >> 2;                   // float4 groups per edge
    int idx = blockIdx.x * blockDim.x + threadIdx.x;
    if (idx >= E * gpe) return;
    int e = idx / gpe;
    int g = (idx - e * gpe) << 2;
    int s = src[e];
    int d = dst[e];
    float nrm = dinv[s] * dinv[d];
    const float4 hv = *(const float4*)(h + (size_t)s * f + g);
    float* o = out + (size_t)d * f + g;
    atomicAdd(o + 0, hv.x * nrm);
    atomicAdd(o + 1, hv.y * nrm);
    atomicAdd(o + 2, hv.z * nrm);
    atomicAdd(o + 3, hv.w * nrm);
}

__global__ void relu_kernel(float* __restrict__ x, int n) {
    int i = blockIdx.x * blockDim.x + threadIdx.x;
    if (i < n) x[i] = fmaxf(x[i], 0.0f);
}

__global__ void combine_kernel(const float* __restrict__ la, const float* __restrict__ lb,
                               float* __restrict__ logits, int n) {
    int i = blockIdx.x * blockDim.x + threadIdx.x;
    if (i < n) logits[i] = 0.5f * (la[i] + lb[i]);
}

// ---------------- host orchestration ----------------

static inline int cdiv(int a, int b) { return (a + b - 1) / b; }

static void run_branch(const float* x, const int* edge, const float* W1, const float* b1,
                       const float* W2, const float* b2, float* l_out,
                       float* dinv, float* h, float* xh, float* h2,
                       int Nn, int IN, int HID, int OUT, int E, hipStream_t stream) {
    const int* src = edge;
    const int* dst = edge + E;

    // normalization
    deg_init_kernel<<<cdiv(Nn, 256), 256, 0, stream>>>(dinv, Nn);
    deg_accum_kernel<<<cdiv(E, 256), 256, 0, stream>>>(dst, dinv, E);
    rsqrt_kernel<<<cdiv(Nn, 256), 256, 0, stream>>>(dinv, Nn);

    // layer 1: h = x @ W1 ; xh = aggregate + bias ; relu
    {
        int waves = (Nn / 16) * (HID / 16);
        gemm_wmma_f32<<<cdiv(waves, 8), 256, 0, stream>>>(x, W1, h, Nn, IN, HID);
        init_out_kernel<<<cdiv(Nn * HID, 256), 256, 0, stream>>>(h, dinv, b1, xh, Nn, HID);
        scatter_kernel<<<cdiv(E * (HID / 4), 256), 256, 0, stream>>>(h, dinv, src, dst, xh, E, HID);
        relu_kernel<<<cdiv(Nn * HID, 256), 256, 0, stream>>>(xh, Nn * HID);
    }

    // layer 2: h2 = xh @ W2 ; l_out = aggregate + bias
    {
        int waves = (Nn / 16) * (OUT / 16);
        gemm_wmma_f32<<<cdiv(waves, 8), 256, 0, stream>>>(xh, W2, h2, Nn, HID, OUT);
        init_out_kernel<<<cdiv(Nn * OUT, 256), 256, 0, stream>>>(h2, dinv, b2, l_out, Nn, OUT);
        scatter_kernel<<<cdiv(E * (OUT / 4), 256), 256, 0, stream>>>(h2, dinv, src, dst, l_out, E, OUT);
    }
}

extern "C" void kernel_launch(void* const* d_in, const int* in_sizes, int n_in,
                              void* d_out, int out_size, void* d_ws, size_t ws_size,
                              hipStream_t stream) {
    const float* x      = (const float*)d_in[0];
    const int*   edge_a = (const int*)  d_in[1];
    const int*   edge_b = (const int*)  d_in[2];
    const float* W1a    = (const float*)d_in[3];
    const float* b1a    = (const float*)d_in[4];
    const float* W2a    = (const float*)d_in[5];
    const float* b2a    = (const float*)d_in[6];
    const float* W1b    = (const float*)d_in[7];
    const float* b1b    = (const float*)d_in[8];
    const float* W2b    = (const float*)d_in[9];
    const float* b2b    = (const float*)d_in[10];

    const int HID = in_sizes[4];            // 96
    const int OUT = in_sizes[6];            // 32
    const int IN  = in_sizes[3] / HID;      // 96
    const int Nn  = in_sizes[0] / IN;       // 50000
    const int E   = in_sizes[1] / 2;        // 800000

    // workspace layout (floats): dinv | h | xh | h2
    float* ws   = (float*)d_ws;
    float* dinv = ws;                       ws += Nn;
    float* h    = ws;                       ws += (size_t)Nn * HID;
    float* xh   = ws;                       ws += (size_t)Nn * HID;
    float* h2   = ws;                       ws += (size_t)Nn * OUT;

    // output layout: logits | la | lb (each Nn x OUT)
    float* logits = (float*)d_out;
    float* la     = logits + (size_t)Nn * OUT;
    float* lb     = la     + (size_t)Nn * OUT;

    run_branch(x, edge_a, W1a, b1a, W2a, b2a, la, dinv, h, xh, h2, Nn, IN, HID, OUT, E, stream);
    run_branch(x, edge_b, W1b, b1b, W2b, b2b, lb, dinv, h, xh, h2, Nn, IN, HID, OUT, E, stream);

    combine_kernel<<<cdiv(Nn * OUT, 256), 256, 0, stream>>>(la, lb, logits, Nn * OUT);
}